// GraphOfGraphsModel_49795850830444
// MI455X (gfx1250) — compile-verified
//
#include <hip/hip_runtime.h>
#include <math.h>

typedef __attribute__((ext_vector_type(2))) float v2f;
typedef __attribute__((ext_vector_type(8))) float v8f;

// ---------------------------------------------------------------------------
// small utility kernels
// ---------------------------------------------------------------------------
__global__ __launch_bounds__(256) void fill_f32(float* __restrict__ p, int n, float v) {
    int i = blockIdx.x * 256 + threadIdx.x;
    if (i < n) p[i] = v;
}

// deg[dst] += 1 for every edge (dst row is the second half of the [2,E] array)
__global__ __launch_bounds__(256) void deg_accum(const int* __restrict__ ei, int E,
                                                 float* __restrict__ deg) {
    int e = blockIdx.x * 256 + threadIdx.x;
    if (e < E) atomicAdd(&deg[ei[E + e]], 1.0f);
}

__global__ __launch_bounds__(256) void rsqrt_inplace(float* __restrict__ p, int n) {
    int i = blockIdx.x * 256 + threadIdx.x;
    if (i < n) p[i] = rsqrtf(p[i]);
}

// ---------------------------------------------------------------------------
// H = (RELU? relu(X) : X) @ W   via V_WMMA_F32_16X16X4_F32
//   X: [nrows, K] row-major, W: [K, F] row-major, H: [nrows, F]
//   one wave (32 lanes) computes a 16-row tile across all F columns
// ---------------------------------------------------------------------------
template <int K, int F, bool RELU>
__global__ __launch_bounds__(256)
void gemm_wmma(const float* __restrict__ X, const float* __restrict__ W,
               float* __restrict__ H, int nrows, int ntiles) {
    constexpr int LST = F + 16;          // padded LDS stride (bank-conflict free)
    __shared__ float sW[K * LST];

    const int tid = threadIdx.x;
    // cooperative stage of W into LDS (float4 granularity)
    for (int i = tid; i < (K * F) / 4; i += 256) {
        const int k  = i / (F / 4);
        const int f4 = (i % (F / 4)) * 4;
        const float4 w = *(const float4*)(W + (long)k * F + f4);
        *(float4*)(&sW[k * LST + f4]) = w;
    }
    __syncthreads();

    const int tile = blockIdx.x * 8 + (tid >> 5);   // wave-uniform
    if (tile >= ntiles) return;                      // wave-uniform exit, EXEC stays full

    const int  lane   = tid & 31;
    const int  m      = lane & 15;          // A-matrix row within tile / C column
    const int  koff   = (lane >> 4) * 2;    // K sub-offset: lanes 0-15 -> 0, 16-31 -> 2
    const int  mstore = (lane >> 4) * 8;    // C: lanes 16-31 hold M = r + 8
    long row = (long)tile * 16 + m;
    const long rowL = row < (long)nrows ? row : (long)nrows - 1;  // clamped load row
    const float* __restrict__ xrow = X + rowL * K;

    for (int ct = 0; ct < F / 16; ++ct) {
        const int col = ct * 16 + m;
        v8f acc = {};
#pragma unroll
        for (int k0 = 0; k0 < K; k0 += 4) {
            // A fragment (16x4 f32): lane holds x[row, k0+koff .. k0+koff+1]
            float a0 = xrow[k0 + koff];
            float a1 = xrow[k0 + koff + 1];
            if (RELU) { a0 = fmaxf(a0, 0.0f); a1 = fmaxf(a1, 0.0f); }
            v2f a; a.x = a0; a.y = a1;
            // B fragment (4x16 f32): lane holds W[k0+koff .. +1, col]
            v2f b;
            b.x = sW[(k0 + koff) * LST + col];
            b.y = sW[(k0 + koff + 1) * LST + col];
            acc = __builtin_amdgcn_wmma_f32_16x16x4_f32(
                false, a, false, b, (short)0, acc, false, false);
        }
        // C/D layout: VGPR r -> (M = mstore + r, N = m)
        float* __restrict__ hp = H + ((long)tile * 16 + mstore) * F + col;
#pragma unroll
        for (int r = 0; r < 8; ++r) {
            const long orow = (long)tile * 16 + mstore + r;
            if (orow < (long)nrows) hp[(long)r * F] = acc[r];
        }
    }
}

// ---------------------------------------------------------------------------
// A[n,f] = H[n,f] * dinv[n]^2 + bias[f]   (self-loop term + bias)
// ---------------------------------------------------------------------------
template <int F>
__global__ __launch_bounds__(256)
void init_accum(const float* __restrict__ H, const float* __restrict__ dinv,
                const float* __restrict__ bias, float* __restrict__ A, long total) {
    constexpr int LOG = (F == 128) ? 7 : 6;
    long i = (long)blockIdx.x * 256 + threadIdx.x;
    if (i >= total) return;
    const int node = (int)(i >> LOG);
    const int f    = (int)(i & (F - 1));
    const float di = dinv[node];
    A[i] = H[i] * di * di + bias[f];
}

// ---------------------------------------------------------------------------
// A[dst] += H[src] * dinv[src]*dinv[dst]  — one wave per edge, vector loads,
// per-float global_atomic_add_f32 (L2-resident: h fits in the 192MB L2)
// ---------------------------------------------------------------------------
template <int F>
__global__ __launch_bounds__(256)
void edge_scatter(const float* __restrict__ H, const int* __restrict__ ei, int E,
                  const float* __restrict__ dinv, float* __restrict__ A) {
    const int edge = blockIdx.x * 8 + (threadIdx.x >> 5);
    if (edge >= E) return;
    const int lane = threadIdx.x & 31;
    const int s = ei[edge];
    const int d = ei[E + edge];
    const float nrm = dinv[s] * dinv[d];
    if constexpr (F == 128) {
        const float4 v = *(const float4*)(H + (long)s * F + lane * 4);
        float* ap = A + (long)d * F + lane * 4;
        atomicAdd(ap + 0, v.x * nrm);
        atomicAdd(ap + 1, v.y * nrm);
        atomicAdd(ap + 2, v.z * nrm);
        atomicAdd(ap + 3, v.w * nrm);
    } else {
        const float2 v = *(const float2*)(H + (long)s * F + lane * 2);
        float* ap = A + (long)d * F + lane * 2;
        atomicAdd(ap + 0, v.x * nrm);
        atomicAdd(ap + 1, v.y * nrm);
    }
}

// ---------------------------------------------------------------------------
// log_softmax over 64 features; one wave per node, 2 floats per lane
// ---------------------------------------------------------------------------
__global__ __launch_bounds__(256)
void log_softmax64(const float* __restrict__ X, float* __restrict__ out, int n) {
    const int node = blockIdx.x * 8 + (threadIdx.x >> 5);
    if (node >= n) return;
    const int lane = threadIdx.x & 31;
    const float2 v = *(const float2*)(X + (long)node * 64 + lane * 2);
    float mx = fmaxf(v.x, v.y);
#pragma unroll
    for (int off = 16; off > 0; off >>= 1) mx = fmaxf(mx, __shfl_xor(mx, off, 32));
    float s = expf(v.x - mx) + expf(v.y - mx);
#pragma unroll
    for (int off = 16; off > 0; off >>= 1) s += __shfl_xor(s, off, 32);
    const float lse = mx + logf(s);
    float2 o; o.x = v.x - lse; o.y = v.y - lse;
    *(float2*)(out + (long)node * 64 + lane * 2) = o;
}

// ---------------------------------------------------------------------------
// driver
// ---------------------------------------------------------------------------
static inline int cdiv(long a, long b) { return (int)((a + b - 1) / b); }

extern "C" void kernel_launch(void* const* d_in, const int* in_sizes, int n_in,
                              void* d_out, int out_size, void* d_ws, size_t ws_size,
                              hipStream_t stream) {
    (void)n_in; (void)out_size; (void)ws_size;
    const float* x  = (const float*)d_in[0];   // [N, 64]
    const int*   ei = (const int*)d_in[1];     // [2, E1]
    const int*   mi = (const int*)d_in[2];     // [2, E2]
    const float* W1 = (const float*)d_in[3];   // [64, 128]
    const float* b1 = (const float*)d_in[4];   // [128]
    const float* W2 = (const float*)d_in[5];   // [128, 64]
    const float* b2 = (const float*)d_in[6];   // [64]

    const int N  = in_sizes[0] / 64;
    const int E1 = in_sizes[1] / 2;
    const int E2 = in_sizes[2] / 2;

    float* ws    = (float*)d_ws;
    float* dinvB = ws;                        // [N]
    float* dinvM = dinvB + N;                 // [N]
    float* H     = dinvM + N;                 // [N,128] gemm output
    float* A     = H + (long)N * 128;         // [N,128] accumulator
    float* B     = A + (long)N * 128;         // [N,64]  accumulator
    float* out   = (float*)d_out;             // [N,64]

    // ---- degrees (self-loop baked in as initial 1.0) ----
    fill_f32<<<cdiv(N, 256), 256, 0, stream>>>(dinvB, N, 1.0f);
    fill_f32<<<cdiv(N, 256), 256, 0, stream>>>(dinvM, N, 1.0f);
    deg_accum<<<cdiv(E1, 256), 256, 0, stream>>>(ei, E1, dinvB);
    deg_accum<<<cdiv(E2, 256), 256, 0, stream>>>(mi, E2, dinvM);
    rsqrt_inplace<<<cdiv(N, 256), 256, 0, stream>>>(dinvB, N);
    rsqrt_inplace<<<cdiv(N, 256), 256, 0, stream>>>(dinvM, N);

    const int ntiles = cdiv(N, 16);
    const int gblk   = cdiv(ntiles, 8);

    // ---- conv1 on base graph: A = agg(x @ W1) + b1  (relu deferred) ----
    gemm_wmma<64, 128, false><<<gblk, 256, 0, stream>>>(x, W1, H, N, ntiles);
    init_accum<128><<<cdiv((long)N * 128, 256), 256, 0, stream>>>(H, dinvB, b1, A, (long)N * 128);
    edge_scatter<128><<<cdiv(E1, 8), 256, 0, stream>>>(H, ei, E1, dinvB, A);

    // ---- conv2 on base graph: B = agg(relu(A) @ W2) + b2 ----
    gemm_wmma<128, 64, true><<<gblk, 256, 0, stream>>>(A, W2, H, N, ntiles);
    init_accum<64><<<cdiv((long)N * 64, 256), 256, 0, stream>>>(H, dinvB, b2, B, (long)N * 64);
    edge_scatter<64><<<cdiv(E1, 8), 256, 0, stream>>>(H, ei, E1, dinvB, B);

    // ---- conv1 on meta graph: A = agg(B @ W1) + b1  (relu deferred) ----
    gemm_wmma<64, 128, false><<<gblk, 256, 0, stream>>>(B, W1, H, N, ntiles);
    init_accum<128><<<cdiv((long)N * 128, 256), 256, 0, stream>>>(H, dinvM, b1, A, (long)N * 128);
    edge_scatter<128><<<cdiv(E2, 8), 256, 0, stream>>>(H, mi, E2, dinvM, A);

    // ---- conv2 on meta graph: B = agg(relu(A) @ W2) + b2 ----
    gemm_wmma<128, 64, true><<<gblk, 256, 0, stream>>>(A, W2, H, N, ntiles);
    init_accum<64><<<cdiv((long)N * 64, 256), 256, 0, stream>>>(H, dinvM, b2, B, (long)N * 64);
    edge_scatter<64><<<cdiv(E2, 8), 256, 0, stream>>>(H, mi, E2, dinvM, B);

    // ---- log_softmax ----
    log_softmax64<<<cdiv(N, 8), 256, 0, stream>>>(B, out, N);
}